// ConditionalInstanceNorm2d_56702158241890
// MI455X (gfx1250) — compile-verified
//
#include <hip/hip_runtime.h>
#include <math.h>

typedef __attribute__((ext_vector_type(2))) float v2f;
typedef __attribute__((ext_vector_type(4))) float vf4;
typedef __attribute__((ext_vector_type(8))) float v8f;

#define C_CH   256
#define NROW   10
#define BATCH  16
#define CDIM   5
#define PLANE  16384          // H*W = 128*128
#define MOM    0.1f

// ---------------------------------------------------------------------------
// Kernel 1: per-(b,c) mean / std over the 128x128 plane.
// ---------------------------------------------------------------------------
__global__ __launch_bounds__(256) void cin_reduce(const float* __restrict__ im,
                                                  float* __restrict__ mean_ws,
                                                  float* __restrict__ std_ws) {
    __shared__ float s_sum[256];
    __shared__ float s_sq[256];
    const int bc = blockIdx.x;                         // b*256 + c
    const vf4* p = (const vf4*)(im + (size_t)bc * PLANE);
    float sum = 0.f, sq = 0.f;
#pragma unroll
    for (int k = 0; k < 16; ++k) {
        vf4 v = p[threadIdx.x + k * 256];
        sum += v.x + v.y + v.z + v.w;
        sq  += v.x * v.x + v.y * v.y + v.z * v.z + v.w * v.w;
    }
    s_sum[threadIdx.x] = sum;
    s_sq[threadIdx.x]  = sq;
    __syncthreads();
    for (int off = 128; off > 0; off >>= 1) {
        if (threadIdx.x < off) {
            s_sum[threadIdx.x] += s_sum[threadIdx.x + off];
            s_sq[threadIdx.x]  += s_sq[threadIdx.x + off];
        }
        __syncthreads();
    }
    if (threadIdx.x == 0) {
        const float N = (float)PLANE;
        float mean = s_sum[0] / N;
        float var  = (s_sq[0] - s_sum[0] * mean) / (N - 1.0f);   // unbiased
        mean_ws[bc] = mean;
        std_ws[bc]  = sqrtf(var);
    }
}

// ---------------------------------------------------------------------------
// Kernel 2: serial momentum scan over batch + WMMA matmul (sel @ rs / rm)
// using V_WMMA_F32_16X16X4_F32 in wave 0, fused scale/shift computation.
// ---------------------------------------------------------------------------
__global__ __launch_bounds__(256) void cin_scan_wmma(
        const float* __restrict__ mean_ws, const float* __restrict__ std_ws,
        const float* __restrict__ running_mean, const float* __restrict__ running_std,
        const int* __restrict__ c_trg, const int* __restrict__ c_org,
        float* __restrict__ scale_ws, float* __restrict__ shift_ws) {
    __shared__ float rs_pad[16][C_CH];   // rows 10..15 zero (K padding)
    __shared__ float rm_pad[16][C_CH];

    const int c = threadIdx.x;           // one channel per thread
    float rm[NROW], rs[NROW];
#pragma unroll
    for (int r = 0; r < NROW; ++r) {
        rm[r] = running_mean[r * C_CH + c];
        rs[r] = running_std[r * C_CH + c];
    }
    // Serial scan over batch (EMA compounding: order matters).
    for (int b = 0; b < BATCH; ++b) {
        const float s  = std_ws[b * C_CH + c];
        const float mu = mean_ws[b * C_CH + c];
#pragma unroll
        for (int r = 0; r < NROW; ++r) {
            const int bit = c_org[b * CDIM + ((r < CDIM) ? r : r - CDIM)];
            const bool org = (r < CDIM) ? (bit != 0) : (bit == 0);
            if (org) {
                rs[r] = rs[r] * (1.0f - MOM) + s  * MOM;
                rm[r] = rm[r] * (1.0f - MOM) + mu * MOM;
            }
        }
    }
#pragma unroll
    for (int r = 0; r < NROW; ++r) { rs_pad[r][c] = rs[r]; rm_pad[r][c] = rm[r]; }
#pragma unroll
    for (int r = NROW; r < 16; ++r) { rs_pad[r][c] = 0.f; rm_pad[r][c] = 0.f; }
    __syncthreads();

    // Wave 0 only: all 32 lanes active -> EXEC all ones (WMMA requirement).
    if (threadIdx.x < 32) {
        const int lane  = threadIdx.x;
        const int mrow  = lane & 15;           // M index for A / N index for B
        const int khalf = (lane >> 4) << 1;    // upper lane half holds K rows +2

        // Sel[m][k] = c_trg_full, zero-padded for k in [10,16)
        float selv[16];
#pragma unroll
        for (int k = 0; k < 16; ++k) {
            float v = 0.f;
            if (k < CDIM)            v = (c_trg[mrow * CDIM + k] != 0) ? 1.f : 0.f;
            else if (k < 2 * CDIM)   v = (c_trg[mrow * CDIM + (k - CDIM)] == 0) ? 1.f : 0.f;
            selv[k] = v;
        }

        for (int j = 0; j < 16; ++j) {         // 16 N-tiles covering 256 channels
            const int n = j * 16 + mrow;
            v8f acc_s = {};
            v8f acc_m = {};
#pragma unroll
            for (int base = 0; base < 16; base += 4) {   // K = 16 in steps of 4
                v2f a;                                   // A: 16x4 slice of Sel
                a.x = selv[base + khalf];
                a.y = selv[base + khalf + 1];
                v2f bs, bm;                              // B: 4x16 slice of RS/RM
                const int k0 = base + khalf;
                bs.x = rs_pad[k0][n];     bs.y = rs_pad[k0 + 1][n];
                bm.x = rm_pad[k0][n];     bm.y = rm_pad[k0 + 1][n];
                acc_s = __builtin_amdgcn_wmma_f32_16x16x4_f32(
                            false, a, false, bs, (short)0, acc_s, false, false);
                acc_m = __builtin_amdgcn_wmma_f32_16x16x4_f32(
                            false, a, false, bm, (short)0, acc_m, false, false);
            }
            // D layout: VGPR r -> M = r (lanes 0-15) / r+8 (lanes 16-31)
#pragma unroll
            for (int r = 0; r < 8; ++r) {
                const int mb = r + ((lane >> 4) << 3);   // batch index
                const int cc = j * 16 + mrow;            // channel index
                const float tstd  = acc_s[r] * 0.2f;     // / cnt (cnt == 5 always)
                const float tmean = acc_m[r] * 0.2f;
                const float s  = std_ws[mb * C_CH + cc];
                const float mu = mean_ws[mb * C_CH + cc];
                const float sc = tstd / s;
                scale_ws[mb * C_CH + cc] = sc;
                shift_ws[mb * C_CH + cc] = tmean - mu * sc;
            }
        }
    }
}

// ---------------------------------------------------------------------------
// Kernel 3: streaming apply  out = im*scale[bc] + shift[bc]  (bandwidth-bound,
// non-temporal stores to avoid polluting L2 with the write-once 256 MB output)
// ---------------------------------------------------------------------------
__global__ __launch_bounds__(256) void cin_apply(const float* __restrict__ im,
                                                 const float* __restrict__ scale_ws,
                                                 const float* __restrict__ shift_ws,
                                                 float* __restrict__ out) {
    const int bc    = blockIdx.x >> 2;
    const int chunk = blockIdx.x & 3;
    const float sc = scale_ws[bc];        // uniform per block -> scalar loads
    const float sh = shift_ws[bc];
    const size_t base4 = (size_t)bc * (PLANE / 4) + (size_t)chunk * 1024;
    const vf4* in4 = (const vf4*)im + base4;
    vf4* out4      = (vf4*)out + base4;
#pragma unroll
    for (int k = 0; k < 4; ++k) {
        const int i = threadIdx.x + k * 256;
        vf4 v = __builtin_nontemporal_load(&in4[i]);
        vf4 r;
        r.x = fmaf(v.x, sc, sh);
        r.y = fmaf(v.y, sc, sh);
        r.z = fmaf(v.z, sc, sh);
        r.w = fmaf(v.w, sc, sh);
        __builtin_nontemporal_store(r, &out4[i]);
    }
}

extern "C" void kernel_launch(void* const* d_in, const int* in_sizes, int n_in,
                              void* d_out, int out_size, void* d_ws, size_t ws_size,
                              hipStream_t stream) {
    (void)in_sizes; (void)n_in; (void)out_size; (void)ws_size;
    const float* im           = (const float*)d_in[0];
    const float* running_mean = (const float*)d_in[1];
    const float* running_std  = (const float*)d_in[2];
    const int*   c_trg        = (const int*)d_in[3];
    const int*   c_org        = (const int*)d_in[4];
    float* out = (float*)d_out;

    float* mean_ws  = (float*)d_ws;              // 4096 floats
    float* std_ws   = mean_ws + BATCH * C_CH;    // 4096 floats
    float* scale_ws = std_ws  + BATCH * C_CH;    // 4096 floats
    float* shift_ws = scale_ws + BATCH * C_CH;   // 4096 floats

    cin_reduce<<<BATCH * C_CH, 256, 0, stream>>>(im, mean_ws, std_ws);
    cin_scan_wmma<<<1, 256, 0, stream>>>(mean_ws, std_ws, running_mean,
                                         running_std, c_trg, c_org,
                                         scale_ws, shift_ws);
    cin_apply<<<BATCH * C_CH * 4, 256, 0, stream>>>(im, scale_ws, shift_ws, out);
}